// FullLightGCN_49976239456883
// MI455X (gfx1250) — compile-verified
//
#include <hip/hip_runtime.h>
#include <hip/hip_bf16.h>

#define NUM_USERS 100000
#define NUM_ITEMS 50000
#define N_NODES   150000
#define EMBED     128
#define N_EDGES   1000000
#define HID       128
#define NUM_CAT   100
#define NUM_SKU   2000

typedef float v2f __attribute__((ext_vector_type(2)));
typedef float v8f __attribute__((ext_vector_type(8)));

// ---------------------------------------------------------------- degrees ---
__global__ void deg_kernel(const int* __restrict__ ei, float* __restrict__ deg) {
    int e = blockIdx.x * blockDim.x + threadIdx.x;
    if (e < N_EDGES) {
        atomicAdd(&deg[ei[e]], 1.0f);
        atomicAdd(&deg[ei[N_EDGES + e]], 1.0f);
    }
}

__global__ void dis_kernel(const float* __restrict__ deg, float* __restrict__ dis) {
    int i = blockIdx.x * blockDim.x + threadIdx.x;
    if (i < N_NODES) {
        float d = deg[i];
        dis[i] = (d > 0.0f) ? rsqrtf(d) : 0.0f;
    }
}

// ------------------------------------------------------- graph propagation ---
// One wave per directed edge: 32 lanes x float4 = one 128-float embedding row.
__global__ void prop_kernel(const int* __restrict__ ei, const float* __restrict__ dis,
                            const float* __restrict__ src, float* __restrict__ dst) {
    int gid  = blockIdx.x * blockDim.x + threadIdx.x;
    int t    = gid >> 5;          // wave index == directed-edge index (uniform in wave)
    int lane = gid & 31;
    if (t >= 2 * N_EDGES) return;
    int e = t >> 1, dir = t & 1;
    int r = ei[e];
    int c = ei[N_EDGES + e];
    if (dir) { int tmp = r; r = c; c = tmp; }   // symmetrized reverse edge
    float norm = dis[r] * dis[c];
    const float4* s = (const float4*)(src + (size_t)c * EMBED);
    float4 v = s[lane];
    float* d = dst + (size_t)r * EMBED + lane * 4;
    atomicAdd(d + 0, v.x * norm);
    atomicAdd(d + 1, v.y * norm);
    atomicAdd(d + 2, v.z * norm);
    atomicAdd(d + 3, v.w * norm);
}

__global__ void acc_add_kernel(float* __restrict__ acc, const float* __restrict__ x, int n4) {
    int i = blockIdx.x * blockDim.x + threadIdx.x;
    if (i < n4) {
        float4 a = ((const float4*)acc)[i];
        float4 b = ((const float4*)x)[i];
        a.x += b.x; a.y += b.y; a.z += b.z; a.w += b.w;
        ((float4*)acc)[i] = a;
    }
}

__global__ void scale_kernel(const float* __restrict__ acc, float* __restrict__ u, int n4) {
    int i = blockIdx.x * blockDim.x + threadIdx.x;
    if (i < n4) {
        float4 a = ((const float4*)acc)[i];
        a.x *= 0.25f; a.y *= 0.25f; a.z *= 0.25f; a.w *= 0.25f;
        ((float4*)u)[i] = a;
    }
}

// ------------------------------------------------------------ WMMA GEMM -----
// C[M,N] = act(A[M,128] @ W[128,N] + bias), fp32 via V_WMMA_F32_16X16X4_F32.
// N and ACT are compile-time so every B-fragment load is a global_load_b32
// with an immediate offset (4*k*N*4 bytes, < 24-bit), and the ragged-N lane
// mask is elided when N % 16 == 0. Block = 4 waves; each wave owns one 16x16
// output tile. K=128 -> 32 WMMA per tile. EXEC is all-1s at every WMMA.
template <int N, int ACT /*0=relu,1=sigmoid*/>
__global__ void gemm_k128_kernel(const float* __restrict__ A, const float* __restrict__ W,
                                 const float* __restrict__ bias, float* __restrict__ C) {
    constexpr int nTiles = (N + 15) >> 4;
    const int wave = threadIdx.x >> 5;
    const int lane = threadIdx.x & 31;
    const int nt = blockIdx.y * 4 + wave;
    if (nt >= nTiles) return;                    // wave-uniform

    const int row0 = blockIdx.x * 16;
    const int col0 = nt * 16;
    const int lo   = lane & 15;                  // sub-lane 0..15
    const int hi   = lane >> 4;                  // half-select 0/1
    const int koff = hi * 2;                     // K sub-offset {0,2}

    const int col  = col0 + lo;
    int   colc = col;
    float msk  = 1.0f;
    if constexpr (N % 16 != 0) {
        colc = (col < N) ? col : (N - 1);
        msk  = (col < N) ? 1.0f : 0.0f;
    }

    // A fragment stream: lane holds A[row0+lo][4k+koff .. 4k+koff+1] (aligned float2)
    const float2* Arow  = (const float2*)(A + (size_t)(row0 + lo) * 128);
    // B fragment base: W[koff][colc]; per-k offsets 4*k*N are literals.
    const float*  Wbase = W + koff * N + colc;

    v8f acc = {};
#pragma unroll
    for (int k = 0; k < 32; ++k) {
        float2 af = Arow[2 * k + hi];
        v2f a; a.x = af.x; a.y = af.y;
        float b0 = Wbase[4 * k * N];
        float b1 = Wbase[4 * k * N + N];
        if constexpr (N % 16 != 0) { b0 *= msk; b1 *= msk; }
        v2f b; b.x = b0; b.y = b1;
        acc = __builtin_amdgcn_wmma_f32_16x16x4_f32(
                  false, a, false, b, (short)0, acc, false, false);
    }

    const float bv = bias[colc];
#pragma unroll
    for (int v = 0; v < 8; ++v) {
        float x = acc[v] + bv;
        if constexpr (ACT == 0) x = fmaxf(x, 0.0f);
        else                    x = 1.0f / (1.0f + __expf(-x));
        int row = row0 + v + hi * 8;
        if (col < N) C[(size_t)row * N + col] = x;   // stores after all WMMAs
    }
}

// ---------------------------------------------------------------- launch ----
extern "C" void kernel_launch(void* const* d_in, const int* in_sizes, int n_in,
                              void* d_out, int out_size, void* d_ws, size_t ws_size,
                              hipStream_t stream) {
    const int*   ei       = (const int*)d_in[0];
    const float* user_emb = (const float*)d_in[1];
    const float* item_emb = (const float*)d_in[2];
    const float* churn_w1 = (const float*)d_in[3];
    const float* churn_b1 = (const float*)d_in[4];
    const float* churn_w2 = (const float*)d_in[5];
    const float* churn_b2 = (const float*)d_in[6];
    const float* cat_w1   = (const float*)d_in[7];
    const float* cat_b1   = (const float*)d_in[8];
    const float* cat_w2   = (const float*)d_in[9];
    const float* cat_b2   = (const float*)d_in[10];
    const float* sku_w1   = (const float*)d_in[11];
    const float* sku_b1   = (const float*)d_in[12];
    const float* sku_w2   = (const float*)d_in[13];
    const float* sku_b2   = (const float*)d_in[14];

    float* out       = (float*)d_out;
    float* out_churn = out;                                     // [100k,1]
    float* out_cat   = out_churn + (size_t)NUM_USERS;           // [100k,100]
    float* out_sku   = out_cat   + (size_t)NUM_USERS * NUM_CAT; // [100k,2000]
    float* out_u     = out_sku   + (size_t)NUM_USERS * NUM_SKU; // [100k,128]

    float* ws   = (float*)d_ws;
    float* deg  = ws;
    float* dis  = deg  + N_NODES;
    float* embA = dis  + N_NODES;
    float* embB = embA + (size_t)N_NODES * EMBED;
    float* acc  = embB + (size_t)N_NODES * EMBED;
    float* h    = acc  + (size_t)N_NODES * EMBED;               // [100k,128]

    const size_t embBytes = (size_t)N_NODES * EMBED * sizeof(float);

    // emb = concat(user_emb, item_emb); acc = emb
    hipMemcpyAsync(embA, user_emb, (size_t)NUM_USERS * EMBED * sizeof(float),
                   hipMemcpyDeviceToDevice, stream);
    hipMemcpyAsync(embA + (size_t)NUM_USERS * EMBED, item_emb,
                   (size_t)NUM_ITEMS * EMBED * sizeof(float),
                   hipMemcpyDeviceToDevice, stream);
    hipMemcpyAsync(acc, embA, embBytes, hipMemcpyDeviceToDevice, stream);

    // symmetric degrees -> dis = deg^-0.5
    hipMemsetAsync(deg, 0, (size_t)N_NODES * sizeof(float), stream);
    deg_kernel<<<(N_EDGES + 255) / 256, 256, 0, stream>>>(ei, deg);
    dis_kernel<<<(N_NODES + 255) / 256, 256, 0, stream>>>(deg, dis);

    // 3 propagation layers, acc += layer output
    float* src = embA;
    float* dst = embB;
    const long long propThreads = (long long)2 * N_EDGES * 32;
    const int propBlocks = (int)((propThreads + 255) / 256);
    const int n4 = N_NODES * EMBED / 4;
    for (int l = 0; l < 3; ++l) {
        hipMemsetAsync(dst, 0, embBytes, stream);
        prop_kernel<<<propBlocks, 256, 0, stream>>>(ei, dis, src, dst);
        acc_add_kernel<<<(n4 + 255) / 256, 256, 0, stream>>>(acc, dst, n4);
        float* t = src; src = dst; dst = t;
    }

    // u = acc[:NUM_USERS] / 4  (users are nodes 0..NUM_USERS-1, row-major)
    const int u4 = NUM_USERS * EMBED / 4;
    scale_kernel<<<(u4 + 255) / 256, 256, 0, stream>>>(acc, out_u, u4);

    // MLP heads: hidden = relu(u@W1+b1), out = sigmoid(hidden@W2+b2)
    const int mTiles = NUM_USERS / 16;          // 6250, exact
    dim3 blk(128);
#define GEMM(NN, AA, Ain, Win, Bin, Cout)                                      \
    do {                                                                       \
        constexpr int nT = ((NN) + 15) / 16;                                   \
        dim3 grid(mTiles, (nT + 3) / 4);                                       \
        gemm_k128_kernel<NN, AA><<<grid, blk, 0, stream>>>(Ain, Win, Bin, Cout);\
    } while (0)

    GEMM(HID,     0, out_u, churn_w1, churn_b1, h);
    GEMM(1,       1, h,     churn_w2, churn_b2, out_churn);
    GEMM(HID,     0, out_u, cat_w1,   cat_b1,   h);
    GEMM(NUM_CAT, 1, h,     cat_w2,   cat_b2,   out_cat);
    GEMM(HID,     0, out_u, sku_w1,   sku_b1,   h);
    GEMM(NUM_SKU, 1, h,     sku_w2,   sku_b2,   out_sku);
#undef GEMM
}